// GATLayer_43946105373000
// MI455X (gfx1250) — compile-verified
//
#include <hip/hip_runtime.h>
#include <math.h>

#define N_NODES 50000
#define N_EDGES 800000
#define IN_F    128
#define OUT_F   64

typedef __attribute__((ext_vector_type(2))) float v2f;
typedef __attribute__((ext_vector_type(8))) float v8f;

// ---- workspace layout (float slots) ----
#define WS_H      0                         // N*64 floats
#define WS_SSRC   (N_NODES * OUT_F)         // N floats
#define WS_SDST   (WS_SSRC + N_NODES)       // N floats
#define WS_ALPHA  (WS_SDST + N_NODES)       // E floats
#define WS_ASUM   (WS_ALPHA + N_EDGES)      // N floats
#define WS_GMAX   (WS_ASUM + N_NODES)       // 1 int slot

// Native fp32 global atomic add (global_atomic_add_f32). Data here is
// exp()-weighted sums of normal-range floats, so the "unsafe" denorm/NaN
// caveats do not apply; avoids any CAS-loop fallback in the hot kernel.
__device__ __forceinline__ void atomic_add_f32(float* p, float v) {
  unsafeAtomicAdd(p, v);
}

// ---------------------------------------------------------------------------
// Kernel 1: h = x @ W  via V_WMMA_F32_16X16X4_F32.
// Block = 128 threads (4 wave32s). Block computes a 16x64 tile of h; each
// wave owns one 16x16 output tile and runs K=128 in 32 WMMA steps.
// x-tile and W are staged in LDS (padded strides keep ds_*_b128 aligned and
// the per-iteration A reads bank-conflict free).
// ---------------------------------------------------------------------------
__global__ __launch_bounds__(128) void gat_gemm_wmma(const float* __restrict__ x,
                                                     const float* __restrict__ W,
                                                     float* __restrict__ h) {
  __shared__ __align__(16) float xs[16 * 132];   // 16 rows x 128 K, stride 132
  __shared__ __align__(16) float wsh[128 * 68];  // 128 K x 64 cols, stride 68

  const int tid     = threadIdx.x;
  const int rowBase = blockIdx.x * 16;

  // Stage x tile: 16*128 floats = 512 float4, 4 per thread.
  #pragma unroll
  for (int it = 0; it < 4; ++it) {
    int idx = tid + it * 128;
    int r = idx >> 5, c4 = idx & 31;
    const float4 v = *(const float4*)(x + (rowBase + r) * IN_F + c4 * 4);
    *(float4*)(xs + r * 132 + c4 * 4) = v;
  }
  // Stage W: 128*64 floats = 2048 float4, 16 per thread.
  #pragma unroll
  for (int it = 0; it < 16; ++it) {
    int idx = tid + it * 128;
    int r = idx >> 4, c4 = idx & 15;
    const float4 v = *(const float4*)(W + r * OUT_F + c4 * 4);
    *(float4*)(wsh + r * 68 + c4 * 4) = v;
  }
  __syncthreads();

  const int lane = tid & 31;
  const int wave = tid >> 5;     // 0..3 : which 16-col tile of the 64 outputs
  const int half = lane >> 4;    // 0/1  : K-pair select (A VGPR layout)
  const int l15  = lane & 15;    // A: row M ; B/D: col N
  const int ncol = wave * 16 + l15;

  v8f c = {};
  #pragma unroll
  for (int kk = 0; kk < 32; ++kk) {
    const int k0 = kk * 4 + 2 * half;
    v2f a, b;
    a.x = xs[l15 * 132 + k0];          // A[M=l15][K=k0]
    a.y = xs[l15 * 132 + k0 + 1];      // A[M=l15][K=k0+1]
    b.x = wsh[(k0    ) * 68 + ncol];   // B[K=k0][N=ncol]
    b.y = wsh[(k0 + 1) * 68 + ncol];   // B[K=k0+1][N=ncol]
    c = __builtin_amdgcn_wmma_f32_16x16x4_f32(false, a, false, b,
                                              (short)0, c, false, false);
  }

  // C/D layout: VGPR r -> (M=r, N=lane) for lanes 0-15, (M=r+8, N=lane-16).
  #pragma unroll
  for (int r = 0; r < 8; ++r) {
    h[(rowBase + r + 8 * half) * OUT_F + ncol] = c[r];
  }
}

// ---------------------------------------------------------------------------
// Kernel 2: s_src[n] = h[n]·a[:64], s_dst[n] = h[n]·a[64:]
// ---------------------------------------------------------------------------
__global__ void gat_scores(const float* __restrict__ h, const float* __restrict__ a,
                           float* __restrict__ s_src, float* __restrict__ s_dst) {
  int n = blockIdx.x * blockDim.x + threadIdx.x;
  if (n >= N_NODES) return;
  const float4* h4 = (const float4*)(h + n * OUT_F);
  float ss = 0.f, sd = 0.f;
  #pragma unroll
  for (int i = 0; i < 16; ++i) {
    float4 hv = h4[i];
    float4 as = *(const float4*)(a + i * 4);
    float4 ad = *(const float4*)(a + OUT_F + i * 4);
    ss += hv.x * as.x + hv.y * as.y + hv.z * as.z + hv.w * as.w;
    sd += hv.x * ad.x + hv.y * ad.y + hv.z * ad.z + hv.w * ad.w;
  }
  s_src[n] = ss;
  s_dst[n] = sd;
}

// ---------------------------------------------------------------------------
// Kernel 3: init accumulators (out, alpha_sum, global-max slot).
// ---------------------------------------------------------------------------
__global__ void gat_init(float* __restrict__ out, float* __restrict__ asum,
                         int* __restrict__ gmax) {
  int i = blockIdx.x * blockDim.x + threadIdx.x;
  if (i < N_NODES * OUT_F) out[i] = 0.f;
  if (i < N_NODES)         asum[i] = 0.f;
  if (i == 0)              *gmax = INT_MIN;   // <= ordered encoding of any float
}

// Monotonic float <-> signed-int encoding (involution): preserves order so
// integer atomicMax implements float max including negatives.
__device__ __forceinline__ int ordered_enc(float f) {
  int i = __float_as_int(f);
  return i ^ ((i >> 31) & 0x7fffffff);
}
__device__ __forceinline__ float ordered_dec(int i) {
  return __int_as_float(i ^ ((i >> 31) & 0x7fffffff));
}

// ---------------------------------------------------------------------------
// Kernel 4: per-edge alpha = leaky_relu(s_src[src]+s_dst[dst]) * w; global max.
// ---------------------------------------------------------------------------
__global__ __launch_bounds__(256) void gat_edge_alpha_max(
    const int* __restrict__ src, const int* __restrict__ dst,
    const float* __restrict__ w,
    const float* __restrict__ s_src, const float* __restrict__ s_dst,
    float* __restrict__ alpha, int* __restrict__ gmax) {
  __shared__ float red[256];
  int e = blockIdx.x * 256 + threadIdx.x;
  float al = -INFINITY;
  if (e < N_EDGES) {
    float s = s_src[src[e]] + s_dst[dst[e]];
    s = (s > 0.f) ? s : 0.2f * s;     // leaky_relu, slope 0.2
    al = s * w[e];
    alpha[e] = al;
  }
  red[threadIdx.x] = al;
  __syncthreads();
  #pragma unroll
  for (int off = 128; off > 0; off >>= 1) {
    if (threadIdx.x < off)
      red[threadIdx.x] = fmaxf(red[threadIdx.x], red[threadIdx.x + off]);
    __syncthreads();
  }
  if (threadIdx.x == 0) atomicMax(gmax, ordered_enc(red[0]));
}

// ---------------------------------------------------------------------------
// Kernel 5: alpha_exp = exp(alpha - max); scatter-add denominator and the
// 64-wide numerator sum_e alpha_exp * h[src] into out[dst]. h and out are
// L2-resident (12.8 MB each << 192 MB L2) so the gathers/atomics stay on-chip,
// and unsafeAtomicAdd keeps each update a single L2-side global_atomic_add_f32.
// ---------------------------------------------------------------------------
__global__ __launch_bounds__(256) void gat_edge_scatter(
    const int* __restrict__ src, const int* __restrict__ dst,
    const float* __restrict__ alpha, const int* __restrict__ gmax,
    const float* __restrict__ h,
    float* __restrict__ asum, float* __restrict__ out) {
  int e = blockIdx.x * blockDim.x + threadIdx.x;
  if (e >= N_EDGES) return;
  float gm = ordered_dec(*gmax);
  float ae = __expf(alpha[e] - gm);
  int s = src[e], d = dst[e];
  atomic_add_f32(&asum[d], ae);
  const float4* h4 = (const float4*)(h + s * OUT_F);
  float* o = out + d * OUT_F;
  #pragma unroll
  for (int i = 0; i < 16; ++i) {
    float4 hv = h4[i];
    atomic_add_f32(o + i * 4 + 0, hv.x * ae);
    atomic_add_f32(o + i * 4 + 1, hv.y * ae);
    atomic_add_f32(o + i * 4 + 2, hv.z * ae);
    atomic_add_f32(o + i * 4 + 3, hv.w * ae);
  }
}

// ---------------------------------------------------------------------------
// Kernel 6: out = elu(out / (alpha_sum[dst] + 1e-8))
// ---------------------------------------------------------------------------
__global__ void gat_finalize(const float* __restrict__ asum, float* __restrict__ out) {
  int i = blockIdx.x * blockDim.x + threadIdx.x;
  if (i >= N_NODES * OUT_F) return;
  float denom = asum[i >> 6] + 1e-8f;
  float v = out[i] / denom;
  out[i] = (v > 0.f) ? v : (__expf(v) - 1.f);
}

// ---------------------------------------------------------------------------
extern "C" void kernel_launch(void* const* d_in, const int* in_sizes, int n_in,
                              void* d_out, int out_size, void* d_ws, size_t ws_size,
                              hipStream_t stream) {
  (void)in_sizes; (void)n_in; (void)out_size; (void)ws_size;

  const float* x  = (const float*)d_in[0];
  const int*   ei = (const int*)d_in[1];       // [2, E] flat
  const float* w  = (const float*)d_in[2];
  const float* W  = (const float*)d_in[3];
  const float* a  = (const float*)d_in[4];

  const int* src = ei;
  const int* dst = ei + N_EDGES;

  float* ws    = (float*)d_ws;
  float* h     = ws + WS_H;
  float* s_src = ws + WS_SSRC;
  float* s_dst = ws + WS_SDST;
  float* alpha = ws + WS_ALPHA;
  float* asum  = ws + WS_ASUM;
  int*   gmax  = (int*)(ws + WS_GMAX);
  float* out   = (float*)d_out;

  // 1. h = x @ W (WMMA f32)
  gat_gemm_wmma<<<N_NODES / 16, 128, 0, stream>>>(x, W, h);
  // 2. per-node attention scores
  gat_scores<<<(N_NODES + 255) / 256, 256, 0, stream>>>(h, a, s_src, s_dst);
  // 3. zero accumulators / init global max
  gat_init<<<(N_NODES * OUT_F + 255) / 256, 256, 0, stream>>>(out, asum, gmax);
  // 4. edge scores + global max
  gat_edge_alpha_max<<<(N_EDGES + 255) / 256, 256, 0, stream>>>(
      src, dst, w, s_src, s_dst, alpha, gmax);
  // 5. exp + scatter-add numerator/denominator
  gat_edge_scatter<<<(N_EDGES + 255) / 256, 256, 0, stream>>>(
      src, dst, alpha, gmax, h, asum, out);
  // 6. normalize + ELU
  gat_finalize<<<(N_NODES * OUT_F + 255) / 256, 256, 0, stream>>>(asum, out);
}